// BEMBFlex_5050881540106
// MI455X (gfx1250) — compile-verified
//
#include <hip/hip_runtime.h>
#include <hip/hip_bf16.h>

#define NUM_USERS 100000
#define NUM_ITEMS 25000
#define NUM_CATS  500
#define LATENT    64
#define BATCH     1024
#define KDIM      128   // two concatenated K=64 halves

// LDS B-tile: pair-rows p = k/2 (64 rows), each row = 64 float2 (n) + pad.
// Stride 160 floats => p and p+1 rows are 32 banks apart: lanes 0-15 read
// banks {0..31}, lanes 16-31 read {32..63} -> conflict-free ds_load_b64.
#define BROW 160

typedef __attribute__((ext_vector_type(2))) float v2f;
typedef __attribute__((ext_vector_type(8))) float v8f;

// ---------------------------------------------------------------------------
// Kernel 1: gather user embeddings into packed A matrix [BATCH, 128]
//   Apack[b, 0:64]  = theta_user[user_index[b]]
//   Apack[b,64:128] = zeta_user [user_index[b]]
// ---------------------------------------------------------------------------
__global__ __launch_bounds__(128) void bemb_gather(const float* __restrict__ theta,
                                                   const float* __restrict__ zeta,
                                                   const int*   __restrict__ uidx,
                                                   float*       __restrict__ Apack) {
    const int b = blockIdx.x;
    const int t = threadIdx.x;           // 0..127
    const int u = uidx[b];
    float v = (t < LATENT) ? theta[(size_t)u * LATENT + t]
                           : zeta [(size_t)u * LATENT + (t - LATENT)];
    Apack[(size_t)b * KDIM + t] = v;
}

// ---------------------------------------------------------------------------
// Kernel 2: util[b,i] = lambda[i] + Apack[b,:] . B[i,:]   (B = [alpha‖item_obs])
// FP32 WMMA 16x16x4. Block = 256 threads = 8 waves.
//   grid.x: N tiles of 64 items   grid.y: M blocks of 128 batch rows
//   wave w handles rows [mBase, mBase+16) x 64 item columns (4 WMMA tiles).
// B tile staged in LDS pre-paired: ldsB[p*BROW + 2n] = {B[2p][n], B[2p+1][n]}
// so each WMMA B fragment is one conflict-free ds_load_b64.
// ---------------------------------------------------------------------------
__global__ __launch_bounds__(256) void bemb_gemm(const float* __restrict__ Apack,
                                                 const float* __restrict__ alpha,
                                                 const float* __restrict__ iobs,
                                                 const float* __restrict__ lam,
                                                 float*       __restrict__ out) {
    __shared__ float ldsB[64 * BROW];    // 40 KB

    const int nBase = blockIdx.x * 64;

    // --- cooperative fill: thread t -> column n = t%64, k-range (t/64)*32 ---
    {
        const int t      = threadIdx.x;
        const int n      = t & 63;
        const int kStart = (t >> 6) * 32;            // 0,32,64,96 (never straddles halves)
        const int gi     = nBase + n;
        if (gi < NUM_ITEMS) {
            const float* src = (kStart < LATENT)
                                   ? (alpha + (size_t)gi * LATENT + kStart)
                                   : (iobs  + (size_t)gi * LATENT + (kStart - LATENT));
#pragma unroll
            for (int kk = 0; kk < 32; kk += 4) {
                const float4 v = *(const float4*)(src + kk);
                const int p0 = (kStart + kk) >> 1;   // pair index
                v2f lo; lo.x = v.x; lo.y = v.y;
                v2f hi; hi.x = v.z; hi.y = v.w;
                *(v2f*)&ldsB[(p0    ) * BROW + 2 * n] = lo;
                *(v2f*)&ldsB[(p0 + 1) * BROW + 2 * n] = hi;
            }
        } else {
            v2f z; z.x = 0.f; z.y = 0.f;
#pragma unroll
            for (int kk = 0; kk < 32; kk += 2)
                *(v2f*)&ldsB[((kStart + kk) >> 1) * BROW + 2 * n] = z;
        }
    }
    __syncthreads();

    const int lane  = threadIdx.x & 31;
    const int wv    = threadIdx.x >> 5;              // 0..7
    const int half  = lane >> 4;                     // 0: lanes 0-15, 1: lanes 16-31
    const int nl    = lane & 15;
    const int mBase = blockIdx.y * 128 + wv * 16;

    // --- preload full A fragment for this wave's 16 rows (ISA A-layout 16x4 f32):
    //     lane L<16: VGPR0=K, VGPR1=K+1 ; lane L>=16: K+2, K+3 ; M = lane&15 ---
    const float* aRow = Apack + (size_t)(mBase + nl) * KDIM;
    v2f aF[32];
#pragma unroll
    for (int s = 0; s < 32; ++s)
        aF[s] = *(const v2f*)(aRow + 4 * s + 2 * half);

    v8f acc0 = {0.f,0.f,0.f,0.f,0.f,0.f,0.f,0.f};
    v8f acc1 = acc0, acc2 = acc0, acc3 = acc0;

    // --- K loop: 32 steps of K=4, 4 N-tiles each -> 128 WMMA / wave ---
#pragma unroll
    for (int s = 0; s < 32; ++s) {
        // B fragment rows this lane supplies: K = 2*p, 2*p+1 with p = 2s + half
        const float* bRow = ldsB + (2 * s + half) * BROW;
        const v2f b0 = *(const v2f*)(bRow + 2 * (nl     ));
        const v2f b1 = *(const v2f*)(bRow + 2 * (nl + 16));
        const v2f b2 = *(const v2f*)(bRow + 2 * (nl + 32));
        const v2f b3 = *(const v2f*)(bRow + 2 * (nl + 48));

        acc0 = __builtin_amdgcn_wmma_f32_16x16x4_f32(false, aF[s], false, b0,
                                                     (short)0, acc0, false, false);
        acc1 = __builtin_amdgcn_wmma_f32_16x16x4_f32(false, aF[s], false, b1,
                                                     (short)0, acc1, false, false);
        acc2 = __builtin_amdgcn_wmma_f32_16x16x4_f32(false, aF[s], false, b2,
                                                     (short)0, acc2, false, false);
        acc3 = __builtin_amdgcn_wmma_f32_16x16x4_f32(false, aF[s], false, b3,
                                                     (short)0, acc3, false, false);
    }

    // --- epilogue: C/D layout (VGPR r -> row r + 8*half, lane -> col lane&15) ---
#pragma unroll
    for (int r = 0; r < 8; ++r) {
        const int    m      = mBase + r + 8 * half;
        const size_t rowOff = (size_t)m * NUM_ITEMS;
        int n = nBase + nl;
        if (n < NUM_ITEMS) out[rowOff + n] = acc0[r] + lam[n];
        n += 16; if (n < NUM_ITEMS) out[rowOff + n] = acc1[r] + lam[n];
        n += 16; if (n < NUM_ITEMS) out[rowOff + n] = acc2[r] + lam[n];
        n += 16; if (n < NUM_ITEMS) out[rowOff + n] = acc3[r] + lam[n];
    }
}

// ---------------------------------------------------------------------------
// Kernel 3: per-(category, batch-row) logsumexp. One wave per batch row.
// float-max via order-preserving uint encoding + LDS atomicMax; sum via
// LDS float atomicAdd (ds_add_f32). L2-resident sweeps over util row.
// ---------------------------------------------------------------------------
__device__ __forceinline__ unsigned enc_f32(float f) {
    unsigned u = __float_as_uint(f);
    return (u & 0x80000000u) ? ~u : (u | 0x80000000u);
}
__device__ __forceinline__ float dec_f32(unsigned e) {
    return __uint_as_float((e & 0x80000000u) ? (e ^ 0x80000000u) : ~e);
}

__global__ __launch_bounds__(256) void bemb_reduce(const float* __restrict__ util,
                                                   const int*   __restrict__ cat,
                                                   float*       __restrict__ logZ) {
    __shared__ unsigned mTab[8 * NUM_CATS];   // 16 KB
    __shared__ float    sTab[8 * NUM_CATS];   // 16 KB

    const int lane = threadIdx.x & 31;
    const int wv   = threadIdx.x >> 5;
    const int b    = blockIdx.x * 8 + wv;

    unsigned* mT = mTab + wv * NUM_CATS;
    float*    sT = sTab + wv * NUM_CATS;

    for (int c = lane; c < NUM_CATS; c += 32) { mT[c] = 0x007FFFFFu /* enc(-inf) */; sT[c] = 0.f; }
    __syncthreads();

    const float* row = util + (size_t)b * NUM_ITEMS;

    for (int i = lane; i < NUM_ITEMS; i += 32)
        atomicMax(&mT[cat[i]], enc_f32(row[i]));
    __syncthreads();

    for (int i = lane; i < NUM_ITEMS; i += 32) {
        const int   c = cat[i];
        const float m = dec_f32(mT[c]);
        atomicAdd(&sT[c], __expf(row[i] - m));
    }
    __syncthreads();

    for (int c = lane; c < NUM_CATS; c += 32)
        logZ[(size_t)c * BATCH + b] = dec_f32(mT[c]) + __logf(sT[c]);
}

// ---------------------------------------------------------------------------
// Kernel 4: out[b,i] -= logZ[cat[i]][b]
// ---------------------------------------------------------------------------
__global__ __launch_bounds__(256) void bemb_sub(float*       __restrict__ out,
                                                const int*   __restrict__ cat,
                                                const float* __restrict__ logZ) {
    const size_t idx = (size_t)blockIdx.x * 256 + threadIdx.x;
    if (idx >= (size_t)BATCH * NUM_ITEMS) return;
    const int i = (int)(idx % NUM_ITEMS);
    const int b = (int)(idx / NUM_ITEMS);
    out[idx] -= logZ[(size_t)cat[i] * BATCH + b];
}

// ---------------------------------------------------------------------------
extern "C" void kernel_launch(void* const* d_in, const int* in_sizes, int n_in,
                              void* d_out, int out_size, void* d_ws, size_t ws_size,
                              hipStream_t stream) {
    const float* theta = (const float*)d_in[0];   // [100000,64]
    const float* alpha = (const float*)d_in[1];   // [25000,64]
    const float* zeta  = (const float*)d_in[2];   // [100000,64]
    const float* lam   = (const float*)d_in[3];   // [25000,1]
    const float* iobs  = (const float*)d_in[4];   // [25000,64]
    const int*   uidx  = (const int*)d_in[5];     // [1024]
    const int*   cat   = (const int*)d_in[6];     // [25000]
    float*       out   = (float*)d_out;           // [1024,25000]

    float* Apack = (float*)d_ws;                                  // 1024*128*4  = 512 KB
    float* logZ  = (float*)((char*)d_ws + (size_t)BATCH*KDIM*4);  // 500*1024*4  = 2 MB

    bemb_gather<<<BATCH, 128, 0, stream>>>(theta, zeta, uidx, Apack);

    dim3 gemmGrid((NUM_ITEMS + 63) / 64, BATCH / 128);            // 391 x 8
    bemb_gemm<<<gemmGrid, 256, 0, stream>>>(Apack, alpha, iobs, lam, out);

    bemb_reduce<<<BATCH / 8, 256, 0, stream>>>(out, cat, logZ);

    const size_t total = (size_t)BATCH * NUM_ITEMS;
    bemb_sub<<<(unsigned)((total + 255) / 256), 256, 0, stream>>>(out, cat, logZ);
}